// SDGNN_61306363183323
// MI455X (gfx1250) — compile-verified
//
#include <hip/hip_runtime.h>
#include <hip/hip_bf16.h>
#include <math.h>

#define NN 100000
#define DD 64
#define EE 1600000
#define NEG_SLOPE 0.2f

typedef __attribute__((ext_vector_type(2))) float v2f;
typedef __attribute__((ext_vector_type(8))) float v8f;

// ---------- helpers ----------
__device__ __forceinline__ float leaky(float x) {
    return x > 0.0f ? x : NEG_SLOPE * x;
}
// order-preserving float -> uint encoding (for atomicMax on floats incl. negatives)
__device__ __forceinline__ unsigned enc_f(float f) {
    unsigned u = __float_as_uint(f);
    return (u & 0x80000000u) ? ~u : (u | 0x80000000u);
}
__device__ __forceinline__ float dec_f(unsigned u) {
    return __uint_as_float((u & 0x80000000u) ? (u & 0x7FFFFFFFu) : ~u);
}

// ---------- WMMA f32 GEMM: C[N x 64] = A[N x 64] @ W[64 x 64]  ----------
// acc : 1 -> accumulate into existing C (loads C fragment first)
// bias: optional per-column bias (64 floats), applied at store
// do_tanh: apply tanhf at store
// grid.x = N/32 (N divisible by 32), block = 256 (8 waves; wave -> one 16x16 tile)
__global__ __launch_bounds__(256) void gemm64_wmma(
        const float* __restrict__ A, const float* __restrict__ W,
        float* __restrict__ C, const float* __restrict__ bias,
        int acc, int do_tanh)
{
    __shared__ float sW[64 * 64];
    const int tid = threadIdx.x;

    // stage W into LDS (4096 floats = 1024 float4; 4 per thread)
    {
        const float4* Wv = (const float4*)W;
        float4* sWv = (float4*)sW;
        #pragma unroll
        for (int i = 0; i < 4; ++i) sWv[tid + 256 * i] = Wv[tid + 256 * i];
    }
    __syncthreads();

    const int wave = tid >> 5;           // 0..7
    const int lane = tid & 31;
    const int rowTile = wave >> 2;       // 0..1
    const int colTile = wave & 3;        // 0..3
    const int rowBase = blockIdx.x * 32 + rowTile * 16;
    const int colBase = colTile * 16;
    const int l15 = lane & 15;
    const int hi = lane >> 4;            // 0: K pair {k,k+1}, 1: {k+2,k+3}

    v8f c = {};
    if (acc) {
        #pragma unroll
        for (int i = 0; i < 8; ++i)
            c[i] = C[(rowBase + i + hi * 8) * 64 + colBase + l15];
    }

    const float* Arow = A + (size_t)(rowBase + l15) * 64;
    #pragma unroll
    for (int k = 0; k < 64; k += 4) {
        v2f a, b;
        // A 16x4 f32 fragment: lanes 0-15 hold K={k,k+1}, lanes 16-31 K={k+2,k+3}
        a.x = Arow[k + hi * 2 + 0];
        a.y = Arow[k + hi * 2 + 1];
        // B 4x16 f32 fragment (row-of-K striped across lanes), mirrored K-split
        b.x = sW[(k + hi * 2 + 0) * 64 + colBase + l15];
        b.y = sW[(k + hi * 2 + 1) * 64 + colBase + l15];
        c = __builtin_amdgcn_wmma_f32_16x16x4_f32(false, a, false, b,
                                                  (short)0, c, false, false);
    }

    #pragma unroll
    for (int i = 0; i < 8; ++i) {
        const int r = rowBase + i + hi * 8;
        float v = c[i];
        if (bias)    v += bias[colBase + l15];
        if (do_tanh) v = tanhf(v);
        C[(size_t)r * 64 + colBase + l15] = v;
    }
}

// ---------- per-node attention scores + self-loop init ----------
__global__ void att_scores(const float* __restrict__ h,
                           const float* __restrict__ att_src,
                           const float* __restrict__ att_dst,
                           float* __restrict__ a_src, float* __restrict__ a_dst,
                           float* __restrict__ alpha_self,
                           unsigned* __restrict__ m_enc, int n)
{
    int i = blockIdx.x * blockDim.x + threadIdx.x;
    if (i >= n) return;
    const float4* hv = (const float4*)(h + (size_t)i * 64);
    const float4* as = (const float4*)att_src;
    const float4* ad = (const float4*)att_dst;
    float s0 = 0.f, s1 = 0.f;
    #pragma unroll
    for (int j = 0; j < 16; ++j) {
        float4 v = hv[j], a = as[j], d = ad[j];
        s0 += v.x * a.x + v.y * a.y + v.z * a.z + v.w * a.w;
        s1 += v.x * d.x + v.y * d.y + v.z * d.z + v.w * d.w;
    }
    a_src[i] = s0;
    a_dst[i] = s1;
    float al = leaky(s0 + s1);       // self-loop edge (s=d=i)
    alpha_self[i] = al;
    m_enc[i] = enc_f(al);            // seed segment-max with self-loop
}

// ---------- edge pass 1: segment max ----------
__global__ void edge_max(const int* __restrict__ src, const int* __restrict__ dst,
                         const float* __restrict__ a_src, const float* __restrict__ a_dst,
                         unsigned* __restrict__ m_enc, int e)
{
    int i = blockIdx.x * blockDim.x + threadIdx.x;
    if (i >= e) return;
    int s = src[i], d = dst[i];
    float al = leaky(a_src[s] + a_dst[d]);
    atomicMax(&m_enc[d], enc_f(al));
}

// ---------- decode max, seed z with self-loop ----------
__global__ void seg_mz(const unsigned* __restrict__ m_enc,
                       const float* __restrict__ alpha_self,
                       float* __restrict__ m_f, float* __restrict__ z, int n)
{
    int i = blockIdx.x * blockDim.x + threadIdx.x;
    if (i >= n) return;
    float m = dec_f(m_enc[i]);
    m_f[i] = m;
    z[i] = expf(alpha_self[i] - m);
}

// ---------- edge pass 2: softmax denominator ----------
__global__ void edge_z(const int* __restrict__ src, const int* __restrict__ dst,
                       const float* __restrict__ a_src, const float* __restrict__ a_dst,
                       const float* __restrict__ m_f, float* __restrict__ z, int e)
{
    int i = blockIdx.x * blockDim.x + threadIdx.x;
    if (i >= e) return;
    int s = src[i], d = dst[i];
    float al = leaky(a_src[s] + a_dst[d]);
    atomicAdd(&z[d], expf(al - m_f[d]));
}

// ---------- init out with self-loop contribution + bias ----------
// one thread per (node, feature): gid = i*64 + j
__global__ void out_init(const float* __restrict__ h,
                         const float* __restrict__ alpha_self,
                         const float* __restrict__ m_f, const float* __restrict__ z,
                         const float* __restrict__ bias,
                         float* __restrict__ out, int n)
{
    int gid = blockIdx.x * blockDim.x + threadIdx.x;
    if (gid >= n * 64) return;
    int i = gid >> 6;
    int j = gid & 63;
    float w = expf(alpha_self[i] - m_f[i]) / z[i];
    out[gid] = bias[j] + w * h[gid];
}

// ---------- edge pass 3: weighted aggregation (wave per edge) ----------
// each lane handles 2 features (float2): coalesced 256B gather + 256B atomic scatter per edge
__global__ __launch_bounds__(256) void edge_agg(
        const int* __restrict__ src, const int* __restrict__ dst,
        const float* __restrict__ a_src, const float* __restrict__ a_dst,
        const float* __restrict__ m_f, const float* __restrict__ z,
        const float* __restrict__ h, float* __restrict__ out, int e)
{
    int wid = (blockIdx.x * blockDim.x + threadIdx.x) >> 5;
    int lane = threadIdx.x & 31;
    if (wid >= e) return;
    int s = src[wid], d = dst[wid];                 // uniform per wave -> broadcast load
    float al = leaky(a_src[s] + a_dst[d]);
    float w = expf(al - m_f[d]) / z[d];
    float2 hv = ((const float2*)(h + (size_t)s * 64))[lane];
    float* o = out + (size_t)d * 64 + lane * 2;
    atomicAdd(o + 0, w * hv.x);
    atomicAdd(o + 1, w * hv.y);
}

// ---------- orchestration ----------
extern "C" void kernel_launch(void* const* d_in, const int* in_sizes, int n_in,
                              void* d_out, int out_size, void* d_ws, size_t ws_size,
                              hipStream_t stream) {
    const float* x        = (const float*)d_in[0];
    const int*   edges[4] = {(const int*)d_in[1], (const int*)d_in[2],
                             (const int*)d_in[3], (const int*)d_in[4]};
    const float* gat_W    = (const float*)d_in[5];   // [2,4,64,64]
    const float* gat_as   = (const float*)d_in[6];   // [2,4,64]
    const float* gat_ad   = (const float*)d_in[7];   // [2,4,64]
    const float* gat_b    = (const float*)d_in[8];   // [2,4,64]
    const float* mlp_W1   = (const float*)d_in[9];   // [2,320,64]
    const float* mlp_b1   = (const float*)d_in[10];  // [2,64]
    const float* mlp_W2   = (const float*)d_in[11];  // [2,64,64]
    const float* mlp_b2   = (const float*)d_in[12];  // [2,64]

    // workspace carve-up (floats)
    float* ws   = (float*)d_ws;
    float* h    = ws;                       // N*64
    float* gat  = h   + (size_t)NN * 64;    // N*64
    float* hid  = gat + (size_t)NN * 64;    // N*64
    float* xbuf = hid + (size_t)NN * 64;    // N*64
    float* a_src      = xbuf + (size_t)NN * 64;   // N
    float* a_dst      = a_src + NN;               // N
    float* alpha_self = a_dst + NN;               // N
    float* m_f        = alpha_self + NN;          // N
    float* z          = m_f + NN;                 // N
    unsigned* m_enc   = (unsigned*)(z + NN);      // N

    const int gemmGrid  = NN / 32;                  // 3125 (exact)
    const int nodeGrid  = (NN + 255) / 256;         // 391
    const int featGrid  = (NN * 64 + 255) / 256;    // 25000
    const int edgeGrid  = (EE + 255) / 256;         // 6250
    const int aggGrid   = (EE * 32 + 255) / 256;    // 200000 (wave per edge)

    for (int l = 0; l < 2; ++l) {
        const float* xin  = (l == 0) ? x : xbuf;
        float*       xout = (l == 1) ? (float*)d_out : xbuf;
        const float* W1l  = mlp_W1 + (size_t)l * 320 * 64;

        // hidden_pre = xin @ W1[block 0]
        gemm64_wmma<<<gemmGrid, 256, 0, stream>>>(xin, W1l, hid, nullptr, 0, 0);

        for (int r = 0; r < 4; ++r) {
            const int lr = l * 4 + r;
            const int* src = edges[r];
            const int* dst = edges[r] + EE;

            // h = xin @ gat_W[l][r]
            gemm64_wmma<<<gemmGrid, 256, 0, stream>>>(
                xin, gat_W + (size_t)lr * 64 * 64, h, nullptr, 0, 0);

            att_scores<<<nodeGrid, 256, 0, stream>>>(
                h, gat_as + lr * 64, gat_ad + lr * 64,
                a_src, a_dst, alpha_self, m_enc, NN);

            edge_max<<<edgeGrid, 256, 0, stream>>>(src, dst, a_src, a_dst, m_enc, EE);

            seg_mz<<<nodeGrid, 256, 0, stream>>>(m_enc, alpha_self, m_f, z, NN);

            edge_z<<<edgeGrid, 256, 0, stream>>>(src, dst, a_src, a_dst, m_f, z, EE);

            out_init<<<featGrid, 256, 0, stream>>>(
                h, alpha_self, m_f, z, gat_b + lr * 64, gat, NN);

            edge_agg<<<aggGrid, 256, 0, stream>>>(
                src, dst, a_src, a_dst, m_f, z, h, gat, EE);

            // hidden_pre += gat @ W1[block r+1]; last block fuses +b1 and tanh
            gemm64_wmma<<<gemmGrid, 256, 0, stream>>>(
                gat, W1l + (size_t)(r + 1) * 64 * 64, hid,
                (r == 3) ? (mlp_b1 + l * 64) : nullptr, 1, (r == 3) ? 1 : 0);
        }

        // layer output = hidden @ W2 + b2
        gemm64_wmma<<<gemmGrid, 256, 0, stream>>>(
            hid, mlp_W2 + (size_t)l * 64 * 64, xout, mlp_b2 + l * 64, 0, 0);
    }
}